// FourLayerGCN_3728031613396
// MI455X (gfx1250) — compile-verified
//
#include <hip/hip_runtime.h>
#include <hip/hip_bf16.h>

typedef __attribute__((ext_vector_type(2))) float v2f;
typedef __attribute__((ext_vector_type(8))) float v8f;

// ---------------------------------------------------------------------------
// Utility kernels
// ---------------------------------------------------------------------------
__global__ void fill_f32_kernel(float* __restrict__ p, float v, long long n) {
  long long i = (long long)blockIdx.x * blockDim.x + threadIdx.x;
  if (i < n) p[i] = v;
}

// Zero-pad X[N,34] -> XP[N,36] so the GEMM A panel needs no bounds guards.
__global__ void pad_x_kernel(const float* __restrict__ X, float* __restrict__ XP,
                             long long n36, long long n34total) {
  long long i = (long long)blockIdx.x * blockDim.x + threadIdx.x;
  if (i >= n36) return;
  long long node = i / 36;
  int c = (int)(i - node * 36);
  XP[i] = (c < 34) ? X[node * 34 + c] : 0.0f;
  (void)n34total;
}

// Convert int64 edge list -> int32 (halves index traffic; fits L2),
// and accumulate destination degree (deg pre-initialized to 1.0 = self loop).
__global__ void prep_edges_kernel(const long long* __restrict__ e64,
                                  int* __restrict__ src32, int* __restrict__ dst32,
                                  float* __restrict__ deg, long long E) {
  long long i = (long long)blockIdx.x * blockDim.x + threadIdx.x;
  if (i >= E) return;
  int s = (int)e64[i];
  int d = (int)e64[E + i];
  src32[i] = s;
  dst32[i] = d;
  atomicAdd(&deg[d], 1.0f);
}

// deg -> 1/sqrt(deg) in place
__global__ void dinv_kernel(float* __restrict__ degdinv, int n) {
  int i = blockIdx.x * blockDim.x + threadIdx.x;
  if (i < n) {
    float dg = degdinv[i];
    degdinv[i] = (dg > 0.0f) ? rsqrtf(dg) : 0.0f;
  }
}

// ---------------------------------------------------------------------------
// WMMA GEMM:  H[N,16] = XP[N,KPAD] @ Wpad[KPAD,16]
// XP is pre-padded so KPAD == row stride and KPAD % 4 == 0: A loads are
// unconditional 8-byte vector loads, K loop fully unrolled -> back-to-back
// v_wmma_f32_16x16x4_f32. One wave computes a 16x16 output tile.
//
// f32 WMMA operand layout (ISA 7.12.2):
//   A 16x4 : lanes 0-15 = rows M, lane half selects K pair {0,1} / {2,3}
//   B 4x16 : lanes give column N, VGPR pair gives K (mirrored)
//   C/D    : VGPR r -> row (r + half*8), col = lane&15
// ---------------------------------------------------------------------------
template <int KPAD>
__global__ void gemm16_wmma_kernel(const float* __restrict__ XP, const float* __restrict__ W,
                                   float* __restrict__ H, int n, int din, int dout) {
  __shared__ float Ws[KPAD * 16];                 // [KPAD][16], zero-padded
  for (int i = threadIdx.x; i < KPAD * 16; i += blockDim.x) {
    int k = i >> 4, c = i & 15;
    Ws[i] = (k < din && c < dout) ? W[k * dout + c] : 0.0f;
  }
  __syncthreads();

  int lane = threadIdx.x & 31;
  int wave = threadIdx.x >> 5;
  int row0 = (blockIdx.x * (blockDim.x >> 5) + wave) << 4;   // 16 rows / wave
  if (row0 >= n) return;
  int half = lane >> 4;               // 0: K pair {0,1}; 1: K pair {2,3}
  int l16  = lane & 15;

  const float* xr = XP + (long long)(row0 + l16) * KPAD + half * 2;
  const float* wr = Ws + half * 2 * 16 + l16;

  v8f acc = {};
  #pragma unroll
  for (int k0 = 0; k0 < KPAD; k0 += 4) {
    v2f a = *(const v2f*)(xr + k0);              // global_load_b64, no guard
    v2f b;
    b.x = wr[k0 * 16];                           // ds_load_2addr pair
    b.y = wr[k0 * 16 + 16];
    acc = __builtin_amdgcn_wmma_f32_16x16x4_f32(false, a, false, b,
                                                (short)0, acc, false, false);
  }

  float* hb = H + (long long)(row0 + half * 8) * 16 + l16;
  #pragma unroll
  for (int r = 0; r < 8; ++r) hb[(long long)r * 16] = acc[r];
}

// ---------------------------------------------------------------------------
// Edge scatter: agg[dst] += h[src] * dinv[src] * dinv[dst]
// One thread per edge; float4 gathers, per-DWORD f32 atomics into L2.
// H/AGG row stride fixed at 16; dout4 = dout/4 groups of 4 features.
// ---------------------------------------------------------------------------
__global__ void scatter_kernel(const float* __restrict__ H, const int* __restrict__ src,
                               const int* __restrict__ dst, const float* __restrict__ dinv,
                               float* __restrict__ agg, long long E, int dout4) {
  long long e = (long long)blockIdx.x * blockDim.x + threadIdx.x;
  if (e >= E) return;
  int s = src[e], d = dst[e];
  float w = dinv[s] * dinv[d];
  const float4* hs = (const float4*)(H + (long long)s * 16);
  float* ag = agg + (long long)d * 16;
  #pragma unroll 4
  for (int q = 0; q < dout4; ++q) {
    float4 hv = hs[q];
    atomicAdd(ag + 4 * q + 0, hv.x * w);
    atomicAdd(ag + 4 * q + 1, hv.y * w);
    atomicAdd(ag + 4 * q + 2, hv.z * w);
    atomicAdd(ag + 4 * q + 3, hv.w * w);
  }
}

// ---------------------------------------------------------------------------
// Epilogue: out = agg + h*dinv^2 (folded self-loop) + bias, optional ReLU.
// dout is a power of two (16 or 4) -> shift/mask indexing.
// ---------------------------------------------------------------------------
__global__ void epilogue_kernel(const float* __restrict__ agg, const float* __restrict__ H,
                                const float* __restrict__ dinv, const float* __restrict__ bias,
                                float* __restrict__ out, long long n, int logd, int do_relu) {
  long long i = (long long)blockIdx.x * blockDim.x + threadIdx.x;
  long long node = i >> logd;
  int f = (int)(i & ((1 << logd) - 1));
  if (node >= n) return;
  float di = dinv[node];
  float v = agg[node * 16 + f] + H[node * 16 + f] * (di * di) + bias[f];
  if (do_relu) v = fmaxf(v, 0.0f);
  out[(node << logd) + f] = v;
}

// ---------------------------------------------------------------------------
// Launch
// ---------------------------------------------------------------------------
static inline unsigned gridFor(long long n, int bs) {
  return (unsigned)((n + bs - 1) / bs);
}

extern "C" void kernel_launch(void* const* d_in, const int* in_sizes, int n_in,
                              void* d_out, int out_size, void* d_ws, size_t ws_size,
                              hipStream_t stream) {
  (void)n_in; (void)out_size; (void)ws_size;
  const float*     x0  = (const float*)d_in[0];
  const long long* e64 = (const long long*)d_in[1];
  const float* Wm[4] = {(const float*)d_in[2], (const float*)d_in[4],
                        (const float*)d_in[6], (const float*)d_in[8]};
  const float* Bv[4] = {(const float*)d_in[3], (const float*)d_in[5],
                        (const float*)d_in[7], (const float*)d_in[9]};

  const int DIN0 = 34;
  const long long N = in_sizes[0] / DIN0;
  const long long E = (long long)in_sizes[1] / 2;

  // Workspace carve-out (~85 MB total -> L2-resident working set on MI455X)
  char* w = (char*)d_ws;
  auto take = [&](long long bytes) -> char* {
    char* p = w;
    w += (bytes + 255) & ~255LL;
    return p;
  };
  int*   src32 = (int*)  take(E * 4);
  int*   dst32 = (int*)  take(E * 4);
  float* dinv  = (float*)take(N * 4);          // deg, then 1/sqrt(deg)
  float* XP    = (float*)take(N * 36 * 4);     // layer-1 input, K padded to 36
  float* H     = (float*)take(N * 16 * 4);     // X @ W (16-col padded)
  float* AGG   = (float*)take(N * 16 * 4);     // scatter accumulator
  float* XB    = (float*)take(N * 16 * 4);     // next-layer activations

  const int BS = 256;

  // Preprocessing: pad X, degree (self-loop folded as initial 1.0), dinv.
  pad_x_kernel<<<gridFor(N * 36, BS), BS, 0, stream>>>(x0, XP, N * 36, N * 34);
  fill_f32_kernel<<<gridFor(N, BS), BS, 0, stream>>>(dinv, 1.0f, N);
  prep_edges_kernel<<<gridFor(E, BS), BS, 0, stream>>>(e64, src32, dst32, dinv, E);
  dinv_kernel<<<gridFor(N, BS), BS, 0, stream>>>(dinv, (int)N);

  const float* xin = XP;
  const int douts[4] = {16, 16, 16, 4};
  const int logds[4] = {4, 4, 4, 2};
  for (int l = 0; l < 4; ++l) {
    int dout = douts[l];
    // 256 threads = 8 waves = 128 output rows per block
    if (l == 0) {
      gemm16_wmma_kernel<36><<<gridFor(N, 128), 256, 0, stream>>>(xin, Wm[l], H,
                                                                  (int)N, 34, dout);
    } else {
      gemm16_wmma_kernel<16><<<gridFor(N, 128), 256, 0, stream>>>(xin, Wm[l], H,
                                                                  (int)N, 16, dout);
    }
    fill_f32_kernel<<<gridFor(N * 16, BS), BS, 0, stream>>>(AGG, 0.0f, N * 16);
    scatter_kernel<<<gridFor(E, BS), BS, 0, stream>>>(H, src32, dst32, dinv, AGG, E, dout / 4);
    float* outp = (l == 3) ? (float*)d_out : XB;
    epilogue_kernel<<<gridFor(N * dout, BS), BS, 0, stream>>>(AGG, H, dinv, Bv[l], outp,
                                                              N, logds[l], (l < 3) ? 1 : 0);
    xin = XB;
  }
}